// MultiAttentionHeads_75952201662751
// MI455X (gfx1250) — compile-verified
//
#include <hip/hip_runtime.h>

#define H_   12
#define HS   64
#define C_   768
#define T_   2048
#define SP   128
#define B_   2
#define M_   (B_ * T_)          // 4096 rows of x flattened over batch

typedef float v2f __attribute__((ext_vector_type(2)));
typedef float v8f __attribute__((ext_vector_type(8)));

// D = A(16x4, f32) * B(4x16, f32) + C(16x16, f32)
__device__ __forceinline__ v8f wmma4(v2f a, v2f b, v8f c) {
    return __builtin_amdgcn_wmma_f32_16x16x4_f32(
        /*neg_a=*/false, a, /*neg_b=*/false, b,
        /*c_mod=*/(short)0, c, /*reuse_a=*/false, /*reuse_b=*/false);
}

// ---------------------------------------------------------------------------
// Prep kernels: transpose weights so WMMA B-fragments ({W[kb][n], W[kb+1][n]})
// become a single contiguous 8-byte (b64) load from Wt[n][kb..kb+1].
// ---------------------------------------------------------------------------
__global__ void transpose_qkv_w(const float* __restrict__ Wq,
                                const float* __restrict__ Wk,
                                const float* __restrict__ Wv,
                                float* __restrict__ WqT,
                                float* __restrict__ WkT,
                                float* __restrict__ WvT) {
    const int N = H_ * C_ * HS;
    for (int i = blockIdx.x * blockDim.x + threadIdx.x; i < N;
         i += gridDim.x * blockDim.x) {
        const int d = i % HS;
        const int c = (i / HS) % C_;
        const int h = i / (HS * C_);
        const size_t dst = ((size_t)h * HS + d) * C_ + c;   // [H][64][768]
        WqT[dst] = Wq[i];
        WkT[dst] = Wk[i];
        WvT[dst] = Wv[i];
    }
}

__global__ void transpose_wp(const float* __restrict__ Wp,
                             float* __restrict__ WpT) {
    const int N = C_ * C_;
    for (int i = blockIdx.x * blockDim.x + threadIdx.x; i < N;
         i += gridDim.x * blockDim.x) {
        const int n = i % C_;
        const int c = i / C_;
        WpT[(size_t)n * C_ + c] = Wp[i];                    // [768][768]^T
    }
}

// ---------------------------------------------------------------------------
// Kernel 1: Q/K/V projections.  One wave computes the 16x64 q, k and v tiles
// for one (16-row tile, head).  Software-pipelined: iteration k's 12 WMMAs
// run on resident registers while iteration k+1's 13 b64 loads are in flight.
// ---------------------------------------------------------------------------
__global__ __launch_bounds__(256) void qkv_kernel(
        const float* __restrict__ x,
        const float* __restrict__ WqT, const float* __restrict__ WkT,
        const float* __restrict__ WvT,
        float* __restrict__ q, float* __restrict__ k, float* __restrict__ v) {
    const int lane = threadIdx.x & 31;
    const int wg   = blockIdx.x * (blockDim.x >> 5) + (threadIdx.x >> 5);
    const int h    = wg % H_;
    const int i0   = (wg / H_) * 16;       // global row tile in [0, 4096)
    const int m    = lane & 15;
    const int half = lane >> 4;

    const float* xrow = x + (size_t)(i0 + m) * C_;
    const float* wqr[4], *wkr[4], *wvr[4];
#pragma unroll
    for (int nt = 0; nt < 4; ++nt) {
        const size_t r = ((size_t)h * HS + nt * 16 + m) * C_;
        wqr[nt] = WqT + r;
        wkr[nt] = WkT + r;
        wvr[nt] = WvT + r;
    }

    v8f aq[4] = {}, ak[4] = {}, av[4] = {};

    // prologue: load fragments for k0 = 0
    v2f a_c, bq_c[4], bk_c[4], bv_c[4];
    {
        const int kb = 2 * half;
        a_c = *(const v2f*)(xrow + kb);
#pragma unroll
        for (int nt = 0; nt < 4; ++nt) {
            bq_c[nt] = *(const v2f*)(wqr[nt] + kb);
            bk_c[nt] = *(const v2f*)(wkr[nt] + kb);
            bv_c[nt] = *(const v2f*)(wvr[nt] + kb);
        }
    }

    for (int k0 = 0; k0 < C_ - 4; k0 += 4) {
        const int kb = k0 + 4 + 2 * half;           // next iteration's K slice
        v2f a_n, bq_n[4], bk_n[4], bv_n[4];
        a_n = *(const v2f*)(xrow + kb);
#pragma unroll
        for (int nt = 0; nt < 4; ++nt) {
            bq_n[nt] = *(const v2f*)(wqr[nt] + kb);
            bk_n[nt] = *(const v2f*)(wkr[nt] + kb);
            bv_n[nt] = *(const v2f*)(wvr[nt] + kb);
        }
#pragma unroll
        for (int nt = 0; nt < 4; ++nt) {
            aq[nt] = wmma4(a_c, bq_c[nt], aq[nt]);
            ak[nt] = wmma4(a_c, bk_c[nt], ak[nt]);
            av[nt] = wmma4(a_c, bv_c[nt], av[nt]);
        }
        a_c = a_n;
#pragma unroll
        for (int nt = 0; nt < 4; ++nt) {
            bq_c[nt] = bq_n[nt];
            bk_c[nt] = bk_n[nt];
            bv_c[nt] = bv_n[nt];
        }
    }
    // epilogue: last K slice
#pragma unroll
    for (int nt = 0; nt < 4; ++nt) {
        aq[nt] = wmma4(a_c, bq_c[nt], aq[nt]);
        ak[nt] = wmma4(a_c, bk_c[nt], ak[nt]);
        av[nt] = wmma4(a_c, bv_c[nt], av[nt]);
    }

#pragma unroll
    for (int nt = 0; nt < 4; ++nt)
#pragma unroll
        for (int e = 0; e < 8; ++e) {
            const int row = e + 8 * half;
            const size_t idx = ((size_t)h * M_ + i0 + row) * HS + nt * 16 + m;
            q[idx] = aq[nt][e];
            k[idx] = ak[nt][e];
            v[idx] = av[nt][e];
        }
}

// ---------------------------------------------------------------------------
// Kernel 2: banded attention.  One wave per (head, batch, 16-row tile).
// blockDim == 32 => single-wave workgroup, __syncthreads lowers to S_NOP.
// ---------------------------------------------------------------------------
__global__ __launch_bounds__(32) void attn_kernel(
        const float* __restrict__ q, const float* __restrict__ k,
        const float* __restrict__ v,
        float* __restrict__ attout,   // [B, T, C] concat-head output
        float* __restrict__ att) {    // [H, B, T, T]
    __shared__ float sc[16][160];     // 16 rows x (<=144 band cols, padded)

    const int lane = threadIdx.x;
    const int it = blockIdx.x & (T_ / 16 - 1);
    const int hb = blockIdx.x >> 7;
    const int b  = hb % B_;
    const int h  = hb / B_;
    const int i0 = it * 16;
    const int jlo   = (i0 >= SP) ? (i0 - SP) : 0;   // 16-aligned
    const int numT  = ((i0 - jlo) >> 4) + 1;        // <= 9 live col tiles
    const int ncols = numT * 16;
    const int m    = lane & 15;
    const int half = lane >> 4;

    const float* qp = q + ((size_t)h * M_ + (size_t)b * T_) * HS;
    const float* kp = k + ((size_t)h * M_ + (size_t)b * T_) * HS;
    const float* vp = v + ((size_t)h * M_ + (size_t)b * T_) * HS;
    const float* qrow = qp + (size_t)(i0 + m) * HS;

    // ---- scores = (q . k^T) * scale, masked, staged to LDS ----
    for (int jt = 0; jt < numT; ++jt) {
        const int j0 = jlo + jt * 16;
        const float* krow = kp + (size_t)(j0 + m) * HS;
        v8f d = {};
#pragma unroll
        for (int k0 = 0; k0 < HS; k0 += 4) {
            const int kb = k0 + 2 * half;
            const v2f a  = *(const v2f*)(qrow + kb);
            const v2f bb = *(const v2f*)(krow + kb);
            d = wmma4(a, bb, d);
        }
#pragma unroll
        for (int e = 0; e < 8; ++e) {
            const int row = e + 8 * half;
            const int i = i0 + row, j = j0 + m;
            float s = d[e] * 0.125f;                 // HEAD_SIZE^-0.5
            if (j > i || (i - j) > SP) s = -1e30f;   // exp() underflows to 0
            sc[row][jt * 16 + m] = s;
        }
    }
    __syncthreads();

    // ---- softmax over the band (one lane per row) ----
    if (lane < 16) {
        float mx = -1e30f;
        for (int c = 0; c < ncols; ++c) mx = fmaxf(mx, sc[lane][c]);
        float sum = 0.0f;
        for (int c = 0; c < ncols; ++c) {
            const float e = expf(sc[lane][c] - mx);
            sc[lane][c] = e;
            sum += e;
        }
        const float inv = 1.0f / sum;
        for (int c = 0; c < ncols; ++c) sc[lane][c] *= inv;
    }
    __syncthreads();

    // ---- out = att @ v  (probs from LDS as the WMMA A operand, b64 reads) ----
    v8f acc[4] = {};
    for (int jt = 0; jt < numT; ++jt) {
#pragma unroll
        for (int k0 = 0; k0 < 16; k0 += 4) {
            const int kb = jt * 16 + k0 + 2 * half;
            const v2f a = *(const v2f*)(&sc[m][kb]);
            const int srow = jlo + kb;
#pragma unroll
            for (int nt = 0; nt < 4; ++nt) {
                const int n = nt * 16 + m;
                v2f bb = { vp[(size_t)srow * HS + n], vp[(size_t)(srow + 1) * HS + n] };
                acc[nt] = wmma4(a, bb, acc[nt]);
            }
        }
    }
#pragma unroll
    for (int nt = 0; nt < 4; ++nt)
#pragma unroll
        for (int e = 0; e < 8; ++e) {
            const int row = e + 8 * half;
            attout[((size_t)b * T_ + i0 + row) * C_ + h * HS + nt * 16 + m] = acc[nt][e];
        }

    // ---- stream full att rows: band from LDS, zeros elsewhere (b128) ----
    const size_t attBase = (((size_t)(h * B_ + b)) * T_ + i0) * T_;
    for (int idx = lane; idx < 16 * (T_ / 4); idx += 32) {
        const int r  = idx / (T_ / 4);
        const int j  = (idx % (T_ / 4)) * 4;
        float4 val = make_float4(0.0f, 0.0f, 0.0f, 0.0f);
        if (j >= jlo && j < jlo + ncols)
            val = *reinterpret_cast<const float4*>(&sc[r][j - jlo]);
        *reinterpret_cast<float4*>(&att[attBase + (size_t)r * T_ + j]) = val;
    }
}

// ---------------------------------------------------------------------------
// Kernel 3: output projection  proj = attout @ Wp + bp   (Wp pre-transposed,
// software-pipelined like kernel 1)
// ---------------------------------------------------------------------------
__global__ __launch_bounds__(256) void proj_kernel(
        const float* __restrict__ attout, const float* __restrict__ WpT,
        const float* __restrict__ bp, float* __restrict__ out) {
    const int lane = threadIdx.x & 31;
    const int wg   = blockIdx.x * (blockDim.x >> 5) + (threadIdx.x >> 5);
    const int n0   = (wg % (C_ / 64)) * 64;
    const int i0   = (wg / (C_ / 64)) * 16;
    const int m    = lane & 15;
    const int half = lane >> 4;

    const float* arow = attout + (size_t)(i0 + m) * C_;
    const float* wpr[4];
#pragma unroll
    for (int nt = 0; nt < 4; ++nt)
        wpr[nt] = WpT + (size_t)(n0 + nt * 16 + m) * C_;

    v8f acc[4] = {};

    v2f a_c, b_c[4];
    {
        const int kb = 2 * half;
        a_c = *(const v2f*)(arow + kb);
#pragma unroll
        for (int nt = 0; nt < 4; ++nt)
            b_c[nt] = *(const v2f*)(wpr[nt] + kb);
    }

    for (int k0 = 0; k0 < C_ - 4; k0 += 4) {
        const int kb = k0 + 4 + 2 * half;
        v2f a_n, b_n[4];
        a_n = *(const v2f*)(arow + kb);
#pragma unroll
        for (int nt = 0; nt < 4; ++nt)
            b_n[nt] = *(const v2f*)(wpr[nt] + kb);
#pragma unroll
        for (int nt = 0; nt < 4; ++nt)
            acc[nt] = wmma4(a_c, b_c[nt], acc[nt]);
        a_c = a_n;
#pragma unroll
        for (int nt = 0; nt < 4; ++nt)
            b_c[nt] = b_n[nt];
    }
#pragma unroll
    for (int nt = 0; nt < 4; ++nt)
        acc[nt] = wmma4(a_c, b_c[nt], acc[nt]);

#pragma unroll
    for (int nt = 0; nt < 4; ++nt) {
        const int n = n0 + nt * 16 + m;
        const float bias = bp[n];
#pragma unroll
        for (int e = 0; e < 8; ++e) {
            const int row = e + 8 * half;
            out[(size_t)(i0 + row) * C_ + n] = acc[nt][e] + bias;
        }
    }
}

// ---------------------------------------------------------------------------
extern "C" void kernel_launch(void* const* d_in, const int* in_sizes, int n_in,
                              void* d_out, int out_size, void* d_ws, size_t ws_size,
                              hipStream_t stream) {
    (void)in_sizes; (void)n_in; (void)out_size; (void)ws_size;
    const float* x  = (const float*)d_in[0];
    const float* Wq = (const float*)d_in[1];
    const float* Wk = (const float*)d_in[2];
    const float* Wv = (const float*)d_in[3];
    const float* Wp = (const float*)d_in[4];
    const float* bp = (const float*)d_in[5];

    float* outp = (float*)d_out;
    float* ws   = (float*)d_ws;

    const size_t QKV  = (size_t)H_ * M_ * HS;   // 3,145,728 floats each
    const size_t WSZ  = (size_t)H_ * C_ * HS;   //   589,824 floats each
    float* q   = ws;
    float* k   = ws + QKV;
    float* v   = ws + 2 * QKV;
    float* ao  = ws + 3 * QKV;                  // [B,T,C] concat-head
    float* WqT = ws + 4 * QKV;
    float* WkT = WqT + WSZ;
    float* WvT = WkT + WSZ;
    float* WpT = WvT + WSZ;                     // 589,824 floats

    float* proj = outp;                         // output 0: [B,T,C]
    float* att  = outp + (size_t)B_ * T_ * C_;  // output 1: [H,B,T,T]

    transpose_qkv_w<<<1152, 256, 0, stream>>>(Wq, Wk, Wv, WqT, WkT, WvT);
    transpose_wp   <<<1152, 256, 0, stream>>>(Wp, WpT);

    // 3072 waves each: 256 row-tiles x 12 heads (k1), 12 n-groups (k3)
    qkv_kernel <<<384, 256, 0, stream>>>(x, WqT, WkT, WvT, q, k, v);
    attn_kernel<<<H_ * B_ * (T_ / 16), 32, 0, stream>>>(q, k, v, ao, att);
    proj_kernel<<<384, 256, 0, stream>>>(ao, WpT, bp, proj);
}